// ResidualOnDirichletExtension2D_23708219474533
// MI455X (gfx1250) — compile-verified
//
#include <hip/hip_runtime.h>

// ---------------- problem constants (from reference) ----------------
#define NB   16        // batch
#define NN   512       // grid N
#define NITER 50       // Jacobi iterations
#define RAD  50        // halo radius = NITER
#define TS   92        // output tile (S - 2*RAD)
#define SS   192       // tile-with-halo; SS*SS divisible by 64
#define NTILE 6        // ceil(512/92)
#define NTHREADS 512   // 16 wave32s
#define HALF_CELLS (SS*SS/2)        // 18432 (pairs fill uB exactly: *8B = 147456)
#define HALF_WORDS (HALF_CELLS/32)  // 576

// ---------------- CDNA5 async global->LDS path ----------------
#if defined(__has_builtin)
#  if __has_builtin(__builtin_amdgcn_global_load_async_to_lds_b64)
#    define HAVE_ASYNC_B64 1
#    pragma message("CDNA5: using __builtin_amdgcn_global_load_async_to_lds_b64")
#  else
#    define HAVE_ASYNC_B64 0
#    pragma message("CDNA5: async_to_lds builtin NOT available, using sync fallback")
#  endif
#  if __has_builtin(__builtin_amdgcn_s_wait_asynccnt)
#    define HAVE_WAIT_ASYNC 1
#  else
#    define HAVE_WAIT_ASYNC 0
#  endif
#else
#  define HAVE_ASYNC_B64 0
#  define HAVE_WAIT_ASYNC 0
#endif

typedef int v2i __attribute__((vector_size(8)));   // matches builtin pointee type

__device__ __forceinline__ void async_copy_b64(const void* g, void* l) {
#if HAVE_ASYNC_B64
  __builtin_amdgcn_global_load_async_to_lds_b64(
      (__attribute__((address_space(1))) v2i*)g,   // global src (AS1)
      (__attribute__((address_space(3))) v2i*)l,   // LDS dst (AS3)
      0 /*imm offset*/, 0 /*cpol*/);
#else
  *reinterpret_cast<unsigned long long*>(l) =
      *reinterpret_cast<const unsigned long long*>(g);
#endif
}

__device__ __forceinline__ void wait_async_lds() {
#if HAVE_WAIT_ASYNC
  __builtin_amdgcn_s_wait_asynccnt(0);
#elif HAVE_ASYNC_B64
  asm volatile("s_wait_asynccnt 0" ::: "memory");
#endif
}

// ---------------- fused kernel ----------------
// One workgroup = one 92x92 output tile of one batch image.
// LDS: uA/uB ping-pong (2 * 192*192*4 = 294,912 B) + packed dir-mask (4,608 B)
//    = 299,520 B  -> requires CDNA5's 320 KB per-WGP LDS.
__global__ __launch_bounds__(NTHREADS)
void jacobi_dirichlet_fused(const float* __restrict__ xmix,   // (B,N,N,4)
                            const float* __restrict__ w_back, // (2,1)
                            const float* __restrict__ b_back, // (1,)
                            const float* __restrict__ logit,  // (1,)
                            const float* __restrict__ ymean,  // (1,)
                            const float* __restrict__ ystd,   // (1,)
                            float* __restrict__ out)          // (B,N,N,1)
{
  __shared__ __align__(16) float uA[SS*SS];
  __shared__ __align__(16) float uB[SS*SS];
  __shared__ unsigned dmask[SS*SS/32];

  const int tid = threadIdx.x;
  const int b   = blockIdx.y;
  const int tjx = blockIdx.x % NTILE;
  const int tiy = blockIdx.x / NTILE;
  const int gi0 = tiy*TS - RAD;   // tile-with-halo origin (may be negative)
  const int gj0 = tjx*TS - RAD;
  const float* __restrict__ gb = xmix + (size_t)b*NN*NN*4;

  // ---- init: 2 phases. Stage (geom,bc) float pairs via async global->LDS
  // into uB (exactly fills it), then convert to u0 into uA + packed dir bits.
  for (int phase = 0; phase < 2; ++phase) {
    const int base = phase * HALF_CELLS;
    for (int c = tid; c < HALF_CELLS; c += NTHREADS) {
      const int idx = base + c;
      const int i = idx / SS, j = idx - i*SS;
      const int gi = gi0 + i, gj = gj0 + j;
      if ((unsigned)gi < NN && (unsigned)gj < NN) {
        const float* src = gb + (((size_t)gi*NN + gj) << 2) + 2; // channels {geom,bc}
        async_copy_b64(src, (char*)uB + 8*(size_t)c);
      }
    }
    wait_async_lds();
    __syncthreads();
    // convert: each thread owns whole 32-cell mask words (no atomics needed)
    for (int w = tid; w < HALF_WORDS; w += NTHREADS) {
      unsigned bits = 0u;
      const int cbase = w * 32;
      #pragma unroll 4
      for (int k = 0; k < 32; ++k) {
        const int c = cbase + k;
        const int idx = base + c;
        const int i = idx / SS, j = idx - i*SS;
        const int gi = gi0 + i, gj = gj0 + j;
        const bool inimg = (unsigned)gi < NN && (unsigned)gj < NN;
        float g = 0.f, bcv = 0.f;
        if (inimg) {
          const float2 p = reinterpret_cast<const float2*>(uB)[c];
          g = p.x; bcv = p.y;
        }
        const bool edge = (gi==0) | (gi==NN-1) | (gj==0) | (gj==NN-1);
        const bool dir  = (!inimg) || edge || (g > 0.5f);
        if (dir) bits |= (1u << k);
        uA[idx] = (dir && inimg) ? bcv : 0.f;
      }
      dmask[base/32 + w] = bits;
    }
    __syncthreads();  // staging fully consumed before next phase overwrites uB
  }

  // ---- 50 Jacobi sweeps, fully LDS-resident, float4-vectorized.
  // Exact region at sweep t is [t+1, SS-2-t]^2; we round the column window
  // outward to 4-alignment (extra cells are garbage-but-unread and in-bounds).
  float* cur = uA;
  float* nxt = uB;
  const int tx = tid & 31;
  const int ty = tid >> 5;           // 0..15
  for (int t = 0; t < NITER; ++t) {
    const int lo = t + 1, hi = SS - 2 - t;
    const int jstart = lo & ~3;      // 4-aligned; >= 0
    for (int i = lo + ty; i <= hi; i += 16) {
      const int rowb = i * SS;
      #pragma unroll 2
      for (int j0 = jstart + tx*4; j0 <= hi; j0 += 128) {
        const int idx = rowb + j0;                    // multiple of 4
        const float4 c = *reinterpret_cast<const float4*>(&cur[idx]);
        const float4 u = *reinterpret_cast<const float4*>(&cur[idx - SS]);
        const float4 d = *reinterpret_cast<const float4*>(&cur[idx + SS]);
        const float lf = cur[idx - 1];                // i>=1 -> idx-1 >= SS-1
        const float rt = cur[idx + 4];                // j0<=188 -> in-bounds
        const unsigned mw = dmask[idx >> 5] >> (idx & 31); // 4 bits, no straddle
        float4 r;
        r.x = (mw & 1u) ? c.x : 0.25f * (lf  + c.y + u.x + d.x);
        r.y = (mw & 2u) ? c.y : 0.25f * (c.x + c.z + u.y + d.y);
        r.z = (mw & 4u) ? c.z : 0.25f * (c.y + c.w + u.z + d.z);
        r.w = (mw & 8u) ? c.w : 0.25f * (c.z + rt  + u.w + d.w);
        *reinterpret_cast<float4*>(&nxt[idx]) = r;
      }
    }
    __syncthreads();
    float* tmp = cur; cur = nxt; nxt = tmp;
  }
  // 50 swaps: final field is in cur (== uA).

  // ---- fused epilogue: e_enc + w_outer*fluid*res_scale*(x_norm . w_back + b)
  const float w0 = w_back[0], w1 = w_back[1], bb = b_back[0];
  const float ym = ymean[0];
  const float inv = 1.f / (ystd[0] + 1e-5f);
  const float rs  = 0.25f / (1.f + __expf(-logit[0]));
  const float scale = 1.f / (float)(NN - 1);
  float* __restrict__ outb = out + (size_t)b*NN*NN;

  for (int i = RAD + ty; i < RAD + TS; i += 16) {
    const int gi = gi0 + i;                 // = tiy*TS + (i-RAD) >= 0
    if (gi >= NN) break;
    const float xs = gi * scale;
    const float dx = fminf(xs, 1.f - xs);
    for (int j = RAD + tx; j < RAD + TS; j += 32) {
      const int gj = gj0 + j;
      if (gj >= NN) continue;
      const int idx = i*SS + j;
      const float e = (cur[idx] - ym) * inv;
      const unsigned bit = (dmask[idx >> 5] >> (idx & 31)) & 1u;
      const float ysf = gj * scale;
      const float d  = fminf(dx, fminf(ysf, 1.f - ysf));
      const float wo = fminf(fmaxf(d * 20.f, 0.f), 1.f);   // clip(d/0.05,0,1)
      const float wf = bit ? 0.f : wo;      // edge cells have wo==0 -> exact
      const float2 xn = *reinterpret_cast<const float2*>(
          gb + (((size_t)gi*NN + gj) << 2));
      const float r = fmaf(xn.x, w0, fmaf(xn.y, w1, bb));
      outb[(size_t)gi*NN + gj] = fmaf(wf * rs, r, e);
    }
  }
}

// ---------------- launch ----------------
extern "C" void kernel_launch(void* const* d_in, const int* in_sizes, int n_in,
                              void* d_out, int out_size, void* d_ws, size_t ws_size,
                              hipStream_t stream) {
  (void)in_sizes; (void)n_in; (void)out_size; (void)d_ws; (void)ws_size;
  const float* xmix = (const float*)d_in[0];
  const float* wb   = (const float*)d_in[1];
  const float* bb   = (const float*)d_in[2];
  const float* lg   = (const float*)d_in[3];
  const float* ym   = (const float*)d_in[4];
  const float* ys   = (const float*)d_in[5];
  dim3 grid(NTILE*NTILE, NB, 1);   // 36 tiles x 16 batches = 576 workgroups
  jacobi_dirichlet_fused<<<grid, NTHREADS, 0, stream>>>(
      xmix, wb, bb, lg, ym, ys, (float*)d_out);
}